// Model_17111149707387
// MI455X (gfx1250) — compile-verified
//
#include <hip/hip_runtime.h>
#include <hip/hip_bf16.h>

// Problem geometry (from setup_inputs): B=2, C=4, N=1e6
static constexpr int       Bk   = 2;
static constexpr int       Ck   = 4;
static constexpr long long Nk   = 1000000;
static constexpr int       BCk  = Bk * Ck;            // 8 channels
static constexpr int       Mk   = (int)(Nk / 8);      // 125000 mask bytes per channel
static constexpr int       BLK  = 256;
static constexpr int       NBLK = (Mk + BLK - 1) / BLK; // 489 blocks per channel

// Output offsets (floats) in flat d_out, in reference return order
static constexpr long long O_MEANS = 0;                    // (B,C,3,N) = 24e6
static constexpr long long O_COLOR = 24000000;             // (B,C,3,N)
static constexpr long long O_M2D   = 48000000;             // (B,C,2,N)
static constexpr long long O_DEP   = 64000000;             // (B,C,N)
static constexpr long long O_RAD   = 72000000;             // (B,C,2,N)
static constexpr long long O_COV   = 88000000;             // (B,C,3,N)
static constexpr long long O_CON   = 112000000;            // (B,C,3,N)
static constexpr long long O_OPA   = 136000000;            // (B,C,N)
static constexpr long long O_FILT  = 144000000;            // (B,C,M)
static constexpr long long O_CNT   = 145000000;            // (B,C)

// ---------------------------------------------------------------------------
// Kernel A: compute mask byte per 8 elements, write filter output,
//           per-block popcount sums.
// grid = (NBLK, BC), block = 256
// ---------------------------------------------------------------------------
__global__ void k_mask(const float* __restrict__ det,
                       const float* __restrict__ depths,
                       const float* __restrict__ m2d,
                       const float* __restrict__ rad,
                       float* __restrict__ out,
                       unsigned char* __restrict__ maskbytes,
                       unsigned int* __restrict__ blockcnt)
{
    const int bc = blockIdx.y;
    const int w  = blockIdx.x * BLK + threadIdx.x;
    unsigned int byte = 0u;
    if (w < Mk) {
        const long long n0  = (long long)w * 8;
        const long long cN  = (long long)bc * Nk + n0;       // det / depths
        const long long c2N = (long long)(bc * 2) * Nk + n0; // means2d/radius x-row
        #pragma unroll
        for (int e = 0; e < 8; ++e) {
            const float d  = det[cN + e];
            const float z  = depths[cN + e];
            const float x  = m2d[c2N + e];
            const float y  = m2d[c2N + Nk + e];
            const float rx = rad[c2N + e];
            const float ry = rad[c2N + Nk + e];
            const bool ok = (d > 0.0f) && (z > 0.0f) && (z < 2.0f) &&
                            (x + rx > 0.0f)   && (x - rx < 200.0f) &&
                            (y + ry > 0.0f)   && (y - ry < 600.0f);
            byte |= (ok ? 1u : 0u) << e;
        }
        maskbytes[(long long)bc * Mk + w] = (unsigned char)byte;
        out[O_FILT + (long long)bc * Mk + w] = (float)byte;   // filter_uint8 as f32
    }
    __shared__ unsigned int s[BLK];
    s[threadIdx.x] = __popc(byte);
    __syncthreads();
    for (int off = BLK / 2; off > 0; off >>= 1) {
        if (threadIdx.x < off) s[threadIdx.x] += s[threadIdx.x + off];
        __syncthreads();
    }
    if (threadIdx.x == 0) blockcnt[bc * NBLK + blockIdx.x] = s[0];
}

// ---------------------------------------------------------------------------
// Kernel B: per-channel exclusive scan of block counts (NBLK=489 <= 512),
//           emit block offsets + cnt (int for later kernels, float to d_out)
// grid = BC, block = 512
// ---------------------------------------------------------------------------
__global__ void k_scan(const unsigned int* __restrict__ blockcnt,
                       unsigned int* __restrict__ blockoff,
                       int* __restrict__ cnt_i,
                       float* __restrict__ out)
{
    const int bc = blockIdx.x;
    const int t  = threadIdx.x;
    __shared__ unsigned int s[512];
    const unsigned int v = (t < NBLK) ? blockcnt[bc * NBLK + t] : 0u;
    s[t] = v;
    __syncthreads();
    for (int off = 1; off < 512; off <<= 1) {
        const unsigned int add = (t >= off) ? s[t - off] : 0u;
        __syncthreads();
        s[t] += add;
        __syncthreads();
    }
    const unsigned int incl = s[t];
    if (t < NBLK) blockoff[bc * NBLK + t] = incl - v;  // exclusive
    if (t == 511) {
        cnt_i[bc] = (int)incl;
        out[O_CNT + bc] = (float)incl;
    }
}

// ---------------------------------------------------------------------------
// Kernel C: fill tail (n >= cnt) of every compacted row with 1.0
// grid = (ceil(N/256), 18, BC), block = 256
// row j in [0,18): region r, component k via static tables
// ---------------------------------------------------------------------------
__global__ void k_fill(const int* __restrict__ cnt_i, float* __restrict__ out)
{
    const long long bases[8] = {O_MEANS, O_COLOR, O_M2D, O_DEP, O_RAD, O_COV, O_CON, O_OPA};
    const int comps[8] = {3, 3, 2, 1, 2, 3, 3, 1};
    const int pre[8]   = {0, 3, 6, 8, 9, 11, 13, 16};

    const int bc = blockIdx.z;
    const int j  = blockIdx.y;
    const long long n = (long long)blockIdx.x * blockDim.x + threadIdx.x;
    if (n >= Nk) return;
    if (n < (long long)cnt_i[bc]) return;   // compacted prefix written by scatter

    int r = 0;
    #pragma unroll
    for (int i = 1; i < 8; ++i) if (j >= pre[i]) r = i;
    const int k = j - pre[r];
    out[bases[r] + ((long long)(bc * comps[r] + k)) * Nk + n] = 1.0f;
}

// ---------------------------------------------------------------------------
// Kernel D: stable scatter. Each thread owns one mask byte (8 elements),
// intra-block exclusive scan (LDS) + global block offset gives its stable
// output position; streams 18 channels per surviving element.
// grid = (NBLK, BC), block = 256
// ---------------------------------------------------------------------------
__global__ void k_scatter(const float* __restrict__ means,
                          const float* __restrict__ colors,
                          const float* __restrict__ opac,
                          const float* __restrict__ m2d,
                          const float* __restrict__ depths,
                          const float* __restrict__ rad,
                          const float* __restrict__ conics,
                          const float* __restrict__ covars,
                          const unsigned char* __restrict__ maskbytes,
                          const unsigned int* __restrict__ blockoff,
                          float* __restrict__ out)
{
    const int bc = blockIdx.y;
    const int b  = bc / Ck;
    const int w  = blockIdx.x * BLK + threadIdx.x;

    unsigned int byte = (w < Mk) ? (unsigned int)maskbytes[(long long)bc * Mk + w] : 0u;
    const unsigned int pc = __popc(byte);

    __shared__ unsigned int s[BLK];
    s[threadIdx.x] = pc;
    __syncthreads();
    for (int off = 1; off < BLK; off <<= 1) {
        const unsigned int add = (threadIdx.x >= off) ? s[threadIdx.x - off] : 0u;
        __syncthreads();
        s[threadIdx.x] += add;
        __syncthreads();
    }
    long long pos = (long long)blockoff[bc * NBLK + blockIdx.x] + (s[threadIdx.x] - pc);
    if (!byte) return;

    const long long nb  = (long long)w * 8;
    const long long cN  = (long long)bc * Nk;
    const long long c2N = (long long)(bc * 2) * Nk;
    const long long c3N = (long long)(bc * 3) * Nk;
    const long long b3N = (long long)(b * 3) * Nk;
    const long long bN  = (long long)b * Nk;

    // gfx1250 prefetch of this thread's 32-byte payload chunks (global_prefetch_b8)
    __builtin_prefetch(&depths[cN + nb], 0, 3);
    __builtin_prefetch(&m2d[c2N + nb], 0, 3);
    __builtin_prefetch(&m2d[c2N + Nk + nb], 0, 3);
    __builtin_prefetch(&rad[c2N + nb], 0, 3);
    __builtin_prefetch(&conics[c3N + nb], 0, 3);
    __builtin_prefetch(&covars[c3N + nb], 0, 3);
    __builtin_prefetch(&means[b3N + nb], 0, 3);
    __builtin_prefetch(&colors[b3N + nb], 0, 3);
    __builtin_prefetch(&opac[bN + nb], 0, 3);

    while (byte) {
        const int e = __ffs((int)byte) - 1;
        byte &= byte - 1u;
        const long long n = nb + e;
        const long long p = pos++;

        // means_c / colors_c : broadcast over C
        #pragma unroll
        for (int k = 0; k < 3; ++k) {
            out[O_MEANS + (long long)(bc * 3 + k) * Nk + p] = means[b3N + (long long)k * Nk + n];
            out[O_COLOR + (long long)(bc * 3 + k) * Nk + p] = colors[b3N + (long long)k * Nk + n];
        }
        // means2d_c / radius_c
        #pragma unroll
        for (int k = 0; k < 2; ++k) {
            out[O_M2D + (long long)(bc * 2 + k) * Nk + p] = m2d[c2N + (long long)k * Nk + n];
            out[O_RAD + (long long)(bc * 2 + k) * Nk + p] = rad[c2N + (long long)k * Nk + n];
        }
        // covars2d_c / conics_c
        #pragma unroll
        for (int k = 0; k < 3; ++k) {
            out[O_COV + (long long)(bc * 3 + k) * Nk + p] = covars[c3N + (long long)k * Nk + n];
            out[O_CON + (long long)(bc * 3 + k) * Nk + p] = conics[c3N + (long long)k * Nk + n];
        }
        // depths_c / opac_c
        out[O_DEP + cN + p] = depths[cN + n];
        out[O_OPA + cN + p] = opac[bN + n];
    }
}

// ---------------------------------------------------------------------------
extern "C" void kernel_launch(void* const* d_in, const int* in_sizes, int n_in,
                              void* d_out, int out_size, void* d_ws, size_t ws_size,
                              hipStream_t stream) {
    const float* means   = (const float*)d_in[0];
    const float* colors  = (const float*)d_in[1];
    const float* det     = (const float*)d_in[2];
    const float* opac    = (const float*)d_in[3];
    const float* m2d     = (const float*)d_in[4];
    const float* depths  = (const float*)d_in[5];
    const float* rad     = (const float*)d_in[6];
    const float* conics  = (const float*)d_in[7];
    const float* covars  = (const float*)d_in[8];
    float* out = (float*)d_out;

    // workspace layout
    unsigned char* maskbytes = (unsigned char*)d_ws;                        // BC*M = 1,000,000 B
    unsigned int*  blockcnt  = (unsigned int*)((char*)d_ws + 1000000);      // BC*NBLK u32
    unsigned int*  blockoff  = blockcnt + BCk * NBLK;                       // BC*NBLK u32
    int*           cnt_i     = (int*)(blockoff + BCk * NBLK);               // BC ints

    dim3 gA(NBLK, BCk);
    k_mask<<<gA, BLK, 0, stream>>>(det, depths, m2d, rad, out, maskbytes, blockcnt);

    k_scan<<<BCk, 512, 0, stream>>>(blockcnt, blockoff, cnt_i, out);

    dim3 gC((unsigned)((Nk + BLK - 1) / BLK), 18, BCk);
    k_fill<<<gC, BLK, 0, stream>>>(cnt_i, out);

    dim3 gD(NBLK, BCk);
    k_scatter<<<gD, BLK, 0, stream>>>(means, colors, opac, m2d, depths, rad,
                                      conics, covars, maskbytes, blockoff, out);
}